// AdaptiveCapsule_2911987826878
// MI455X (gfx1250) — compile-verified
//
#include <hip/hip_runtime.h>
#include <math.h>

// ---- problem constants (fixed by reference) ----
#define BATCH 64
#define ICAPS 1024
#define OCAPS 16
#define DDIM  32
#define EDIM  512
#define NTOT  512      // OCAPS * DDIM

typedef __attribute__((ext_vector_type(2))) float v2f;
typedef __attribute__((ext_vector_type(8))) float v8f;

// =====================================================================
// Phase 1: u_hat[b,i,n] = sum_e (W[i,n,e] + bias[n>>5]) * x[b,i,e]
// Per block: one i, one 128-wide N slice; M=64 (batch), K=512.
// WMMA f32 16x16x4, LDS-staged tiles (Kt=32), pitch 36 (conflict-free).
// =====================================================================
__global__ __launch_bounds__(256)
void caps_gemm(const float* __restrict__ x, const float* __restrict__ W,
               const float* __restrict__ bias, float* __restrict__ uhat) {
  const int i     = blockIdx.x;        // in-capsule index
  const int nbase = blockIdx.y * 128;  // N slice base
  const int tid   = threadIdx.x;
  const int lane  = tid & 31;
  const int wave  = tid >> 5;
  const int mbase = (wave & 3) * 16;   // M tile (batch rows)
  const int ng    = wave >> 2;         // N group of 64 within slice

  __shared__ float xs[64 * 36];        //  9216 B
  __shared__ float wsm[128 * 36];      // 18432 B

  v8f acc[4] = {};                     // 4 N-tiles of 16x16 f32

  const int xr = tid >> 3;             // 0..31 (row per pass)
  const int c4 = tid & 7;              // float4 column within 32-wide K tile

  for (int kb = 0; kb < EDIM; kb += 32) {
    // ---- stage x tile: 64 rows x 32 cols (float4 coalesced) ----
#pragma unroll
    for (int p = 0; p < 2; ++p) {
      int r = xr + p * 32;
      size_t fidx = (size_t)r * (ICAPS * EDIM) + (size_t)i * EDIM + kb + c4 * 4;
      float4 vv = *(const float4*)(x + fidx);
      *(float4*)(&xs[r * 36 + c4 * 4]) = vv;
    }
    // ---- stage W tile (+bias): 128 rows x 32 cols ----
#pragma unroll
    for (int p = 0; p < 4; ++p) {
      int r = xr + p * 32;             // 0..127
      int n = nbase + r;
      size_t fidx = (size_t)i * (NTOT * EDIM) + (size_t)n * EDIM + kb + c4 * 4;
      float4 vv = *(const float4*)(W + fidx);
      float bo = bias[n >> 5];
      vv.x += bo; vv.y += bo; vv.z += bo; vv.w += bo;
      *(float4*)(&wsm[r * 36 + c4 * 4]) = vv;
    }
    __syncthreads();

    const int koff = (lane >> 4) << 1;       // lanes 16-31 hold K+2,K+3
    const int mrow = mbase + (lane & 15);
#pragma unroll
    for (int kk = 0; kk < 32; kk += 4) {
      v2f a = *(const v2f*)(&xs[mrow * 36 + kk + koff]);
#pragma unroll
      for (int j = 0; j < 4; ++j) {
        int nr = ng * 64 + j * 16 + (lane & 15);
        v2f bf = *(const v2f*)(&wsm[nr * 36 + kk + koff]);
        acc[j] = __builtin_amdgcn_wmma_f32_16x16x4_f32(
            /*neg_a=*/false, a, /*neg_b=*/false, bf,
            /*c_mod=*/(short)0, acc[j], /*reuse_a=*/false, /*reuse_b=*/false);
      }
    }
    __syncthreads();
  }

  // ---- epilogue: C/D layout -> u_hat[b, i, n] ----
  const int msub = (lane >> 4) << 3;   // lanes 16-31 hold M+8
  const int ncol = lane & 15;
#pragma unroll
  for (int j = 0; j < 4; ++j) {
    int n = nbase + ng * 64 + j * 16 + ncol;
#pragma unroll
    for (int jj = 0; jj < 8; ++jj) {
      int m = mbase + msub + jj;       // batch row
      size_t o = (size_t)m * (ICAPS * NTOT) + (size_t)i * NTOT + n;
      uhat[o] = acc[j][jj];
    }
  }
}

// =====================================================================
// Fused routing pass (one per iteration):
//   iter>0: logits += <u_hat, v>, then attn = softmax(logits) over O
//   iter==0: attn = 1/16 (logits are zero)
//   accumulate s[b,o,d] += attn * u_hat via global f32 atomics
// Grid (B=64, I/128=8), block 256.
// =====================================================================
__global__ __launch_bounds__(256)
void caps_route(const float* __restrict__ uhat, float* __restrict__ logits,
                const float* __restrict__ v_in, float* __restrict__ s_out,
                int iter) {
  const int b     = blockIdx.x;
  const int ibase = blockIdx.y * 128;
  const int t     = threadIdx.x;

  __shared__ float v_sm[NTOT];           // v for this b
  __shared__ float attn_sm[128 * 17];    // pitch 17 to dodge bank conflicts

  if (iter > 0) {
    v_sm[t]       = v_in[b * NTOT + t];
    v_sm[t + 256] = v_in[b * NTOT + t + 256];
  }
  __syncthreads();

  // ---- phase A: per-i logits update + softmax ----
  if (t < 128) {
    const int i = ibase + t;
    const float* u = uhat + ((size_t)b * ICAPS + i) * NTOT;
    float lg[16];
    if (iter > 0) {
      float* L = logits + ((size_t)b * ICAPS + i) * OCAPS;
#pragma unroll
      for (int o = 0; o < OCAPS; ++o) {
        float dot = 0.f;
        const float4* u4 = (const float4*)(u + o * DDIM);
        const float4* v4 = (const float4*)(v_sm + o * DDIM);
#pragma unroll
        for (int d4 = 0; d4 < 8; ++d4) {
          float4 uu = u4[d4]; float4 vv = v4[d4];
          dot += uu.x * vv.x + uu.y * vv.y + uu.z * vv.z + uu.w * vv.w;
        }
        lg[o] = L[o] + dot;
        L[o]  = lg[o];
      }
      float mx = lg[0];
#pragma unroll
      for (int o = 1; o < OCAPS; ++o) mx = fmaxf(mx, lg[o]);
      float e[16]; float sum = 0.f;
#pragma unroll
      for (int o = 0; o < OCAPS; ++o) { e[o] = __expf(lg[o] - mx); sum += e[o]; }
      float inv = 1.f / sum;
#pragma unroll
      for (int o = 0; o < OCAPS; ++o) attn_sm[t * 17 + o] = e[o] * inv;
    } else {
#pragma unroll
      for (int o = 0; o < OCAPS; ++o) attn_sm[t * 17 + o] = 0.0625f;
    }
  }
  __syncthreads();

  // ---- phase B: s partials, coalesced over n = o*32+d ----
  const int n0 = t, n1 = t + 256;
  const int o0 = n0 >> 5, o1 = n1 >> 5;
  float s0 = 0.f, s1 = 0.f;
  const float* ub = uhat + ((size_t)b * ICAPS + ibase) * NTOT;
  for (int il = 0; il < 128; ++il) {
    const float* u = ub + (size_t)il * NTOT;
    s0 += attn_sm[il * 17 + o0] * u[n0];
    s1 += attn_sm[il * 17 + o1] * u[n1];
  }
  atomicAdd(&s_out[b * NTOT + n0], s0);
  atomicAdd(&s_out[b * NTOT + n1], s1);
}

// =====================================================================
// Squash: v = norm/(1+norm^2) * s  per (b,o) over d (wave32 reduction),
// then reset s to 0 for the next accumulation pass.
// 1024 (b,o) pairs -> 1024 waves -> grid 128 x 256.
// =====================================================================
__global__ __launch_bounds__(256)
void caps_squash(float* __restrict__ s, float* __restrict__ vout) {
  const int wv   = (blockIdx.x * blockDim.x + threadIdx.x) >> 5;  // (b,o)
  const int lane = threadIdx.x & 31;
  float sv = s[wv * 32 + lane];
  float ss = sv * sv;
#pragma unroll
  for (int off = 16; off > 0; off >>= 1) ss += __shfl_xor(ss, off, 32);
  float norm = sqrtf(ss);
  float f = norm / (1.0f + ss);          // ss == norm^2
  vout[wv * 32 + lane] = f * sv;
  s[wv * 32 + lane] = 0.f;
}

__global__ void caps_zero(float* __restrict__ p, int n) {
  int idx = blockIdx.x * blockDim.x + threadIdx.x;
  if (idx < n) p[idx] = 0.f;
}

// =====================================================================
extern "C" void kernel_launch(void* const* d_in, const int* in_sizes, int n_in,
                              void* d_out, int out_size, void* d_ws, size_t ws_size,
                              hipStream_t stream) {
  const float* x    = (const float*)d_in[0];  // (B, I, E)
  const float* W    = (const float*)d_in[1];  // (1, I, O, D, E)
  const float* bias = (const float*)d_in[2];  // (1,1,O,1,1)
  float* out = (float*)d_out;                 // (B, O, D) f32

  // workspace layout (floats)
  float* uhat   = (float*)d_ws;                         // 33,554,432
  float* logits = uhat + (size_t)BATCH * ICAPS * NTOT;  //  1,048,576
  float* s      = logits + (size_t)BATCH * ICAPS * OCAPS; //   32,768
  float* v      = s + BATCH * NTOT;                     //     32,768

  // zero logits + s (contiguous regions)
  const int nz = BATCH * ICAPS * OCAPS + BATCH * NTOT;
  caps_zero<<<(nz + 255) / 256, 256, 0, stream>>>(logits, nz);

  // u_hat GEMM: grid (i, N-slice)
  caps_gemm<<<dim3(ICAPS, 4), 256, 0, stream>>>(x, W, bias, uhat);

  // 3 routing iterations
  for (int it = 0; it < 3; ++it) {
    caps_route<<<dim3(BATCH, ICAPS / 128), 256, 0, stream>>>(uhat, logits, v, s, it);
    caps_squash<<<128, 256, 0, stream>>>(s, it == 2 ? out : v);
  }
}